// LSTMTagger_29626684408231
// MI455X (gfx1250) — compile-verified
//
#include <hip/hip_runtime.h>
#include <hip/hip_bf16.h>
#include <math.h>

typedef __attribute__((ext_vector_type(2))) float v2f;
typedef __attribute__((ext_vector_type(8))) float v8f;

#define HID 8
#define NGATES 32   // 4*HID
#define EMB_DIM 6
#define KPAD 8      // K padded for WMMA (2x K=4 slices)
#define VOCAB 9
#define NTAGS 3

#if __has_builtin(__builtin_amdgcn_tanhf)
#define FAST_TANH(x) __builtin_amdgcn_tanhf(x)
#else
#define FAST_TANH(x) tanhf(x)
#endif

// ---------------------------------------------------------------------------
// Kernel 1: x_gates[T,32] = embedding[tok[t]] @ W_ih^T + b_ih + b_hh
// Per wave: one 16(T) x 32(gates) tile via V_WMMA_F32_16X16X4_F32
//   (2 N-tiles x 2 K-slices, K zero-padded 6 -> 8).
// emb / W_ih / bias staged zero-padded in LDS once per block so all WMMA
// operand fetches are unconditional ds_loads (no exec-masked global loads).
//
// A 16x4 f32 layout: lane<16 -> M=lane,    v0:K=0 v1:K=1
//                    lane>=16 -> M=lane-16, v0:K=2 v1:K=3
// B 4x16  f32 layout: lane<16 -> N=lane,    v0:K=0 v1:K=1
//                     lane>=16 -> N=lane-16, v0:K=2 v1:K=3
// C 16x16 f32 layout: VGPR r, lane<16 -> (M=r, N=lane); lane>=16 -> (M=r+8, N=lane-16)
// ---------------------------------------------------------------------------
__global__ void xgates_wmma_kernel(const int* __restrict__ tok,
                                   const float* __restrict__ emb,    // [9,6]
                                   const float* __restrict__ W_ih,   // [32,6]
                                   const float* __restrict__ b_ih,   // [32]
                                   const float* __restrict__ b_hh,   // [32]
                                   float* __restrict__ xg,           // [T,32]
                                   int T)
{
    __shared__ float s_emb[VOCAB * KPAD];    // 72 floats, zero-padded cols 6,7
    __shared__ float s_wih[NGATES * KPAD];   // 256 floats, zero-padded cols 6,7
    __shared__ float s_bias[NGATES];

    const int tid = threadIdx.x;             // 0..127 (4 waves)

    if (tid < VOCAB * KPAD) {
        const int r = tid >> 3, k = tid & 7;
        s_emb[tid] = (k < EMB_DIM) ? emb[r * EMB_DIM + k] : 0.0f;
    }
    for (int i = tid; i < NGATES * KPAD; i += blockDim.x) {
        const int n = i >> 3, k = i & 7;
        s_wih[i] = (k < EMB_DIM) ? W_ih[n * EMB_DIM + k] : 0.0f;
    }
    if (tid < NGATES) s_bias[tid] = b_ih[tid] + b_hh[tid];
    __syncthreads();

    const int lane  = tid & 31;
    const int wave  = (blockIdx.x * blockDim.x + tid) >> 5;
    const int t0    = wave * 16;
    if (t0 >= T) return;

    const int m     = lane & 15;
    const int ka0   = (lane >> 4) * 2;       // 0 or 2
    const int khalf = lane >> 4;

    const int tk = tok[t0 + m];
    const float* er = s_emb + tk * KPAD + ka0;
    const v2f a_lo = { er[0], er[1] };       // K-slice 0..3
    const v2f a_hi = { er[4], er[5] };       // K-slice 4..7 (cols 6,7 are zero)

    #pragma unroll
    for (int nt = 0; nt < 2; ++nt) {
        const int n = nt * 16 + m;           // gate column 0..31
        const float* wr = s_wih + n * KPAD + ka0;
        const v2f b_lo = { wr[0], wr[1] };
        const v2f b_hi = { wr[4], wr[5] };

        v8f c = {};
        c = __builtin_amdgcn_wmma_f32_16x16x4_f32(false, a_lo, false, b_lo,
                                                  (short)0, c, false, false);
        c = __builtin_amdgcn_wmma_f32_16x16x4_f32(false, a_hi, false, b_hi,
                                                  (short)0, c, false, false);

        const float bias = s_bias[n];
        #pragma unroll
        for (int r = 0; r < 8; ++r) {
            const int row = t0 + r + 8 * khalf;      // M = r + 8*khalf
            xg[row * NGATES + n] = c[r] + bias;
        }
    }
}

// ---------------------------------------------------------------------------
// Kernel 2: serial LSTM scan. One wave; lane n owns gate n.
//   gate order: i = lanes 0..7, f = 8..15, g = 16..23, o = 24..31
// W_hh row per lane in registers; h replicated across lanes in 8 VGPRs.
// sigmoid(x) = 0.5*tanh(0.5x)+0.5 -> one hardware tanh per lane for the gate, one for tanh(c), per
// step; xg load double-buffered so VMEM latency hides behind the step chain;
// matvec split into two independent FMA chains to shorten the dep chain.
// ---------------------------------------------------------------------------
__global__ void lstm_scan_kernel(const float* __restrict__ xg,    // [T,32]
                                 const float* __restrict__ W_hh,  // [32,8]
                                 float* __restrict__ hs,          // [T,8]
                                 int T)
{
    const int lane = threadIdx.x;            // 0..31, one wave
    const int j    = lane & 7;

    float w[HID];
    #pragma unroll
    for (int k = 0; k < HID; ++k) w[k] = W_hh[lane * HID + k];

    float hv[HID];
    #pragma unroll
    for (int k = 0; k < HID; ++k) hv[k] = 0.0f;
    float c = 0.0f;

    const bool is_gcell = (lane >= 16) && (lane < 24);

    float gnext = xg[lane];                  // prefetch t = 0
    for (int t = 0; t < T; ++t) {
        const float xgv = gnext;
        const int tn = (t + 1 < T) ? (t + 1) : t;
        gnext = xg[tn * NGATES + lane];      // issue early; hides behind compute

        // two independent 4-deep FMA chains instead of one 8-deep chain
        float pe = __builtin_fmaf(w[0], hv[0],
                   __builtin_fmaf(w[2], hv[2],
                   __builtin_fmaf(w[4], hv[4], w[6] * hv[6])));
        float po = __builtin_fmaf(w[1], hv[1],
                   __builtin_fmaf(w[3], hv[3],
                   __builtin_fmaf(w[5], hv[5], w[7] * hv[7])));
        const float g = xgv + pe + po;

        // branchless: tanh for g-gate lanes, sigmoid (via tanh) otherwise
        const float arg = is_gcell ? g : 0.5f * g;
        const float th  = FAST_TANH(arg);
        const float a   = is_gcell ? th : __builtin_fmaf(0.5f, th, 0.5f);

        const float i_s = __shfl(a, j,      32);
        const float f_s = __shfl(a, j + 8,  32);
        const float g_t = __shfl(a, j + 16, 32);
        const float o_s = __shfl(a, j + 24, 32);

        c = __builtin_fmaf(f_s, c, i_s * g_t);       // identical across lane groups
        const float h = o_s * FAST_TANH(c);          // lane holds h[j]

        #pragma unroll
        for (int k = 0; k < HID; ++k) hv[k] = __shfl(h, k, 32);

        if (lane < HID) hs[t * HID + lane] = h;      // off the critical path
    }
}

// ---------------------------------------------------------------------------
// Kernel 3: logits = hs @ W_out^T + b_out ; stable softmax over 3 tags.
// One thread per timestep; W_out/b_out reads are wave-uniform (scalar cache).
// ---------------------------------------------------------------------------
__global__ void out_softmax_kernel(const float* __restrict__ hs,     // [T,8]
                                   const float* __restrict__ W_out,  // [3,8]
                                   const float* __restrict__ b_out,  // [3]
                                   float* __restrict__ out,          // [T,3]
                                   int T)
{
    const int t = blockIdx.x * blockDim.x + threadIdx.x;
    if (t >= T) return;

    const float4* hp = (const float4*)(hs + t * HID);
    const float4 h0 = hp[0];
    const float4 h1 = hp[1];

    float l[NTAGS];
    #pragma unroll
    for (int tag = 0; tag < NTAGS; ++tag) {
        const float* wr = W_out + tag * HID;
        float s = b_out[tag];
        s = __builtin_fmaf(h0.x, wr[0], s);
        s = __builtin_fmaf(h0.y, wr[1], s);
        s = __builtin_fmaf(h0.z, wr[2], s);
        s = __builtin_fmaf(h0.w, wr[3], s);
        s = __builtin_fmaf(h1.x, wr[4], s);
        s = __builtin_fmaf(h1.y, wr[5], s);
        s = __builtin_fmaf(h1.z, wr[6], s);
        s = __builtin_fmaf(h1.w, wr[7], s);
        l[tag] = s;
    }

    const float mx = fmaxf(l[0], fmaxf(l[1], l[2]));
    const float e0 = __expf(l[0] - mx);
    const float e1 = __expf(l[1] - mx);
    const float e2 = __expf(l[2] - mx);
    const float inv = 1.0f / (e0 + e1 + e2);
    out[t * NTAGS + 0] = e0 * inv;
    out[t * NTAGS + 1] = e1 * inv;
    out[t * NTAGS + 2] = e2 * inv;
}

// ---------------------------------------------------------------------------
extern "C" void kernel_launch(void* const* d_in, const int* in_sizes, int n_in,
                              void* d_out, int out_size, void* d_ws, size_t ws_size,
                              hipStream_t stream)
{
    const int*   tok   = (const int*)  d_in[0];
    const float* emb   = (const float*)d_in[1];
    const float* W_ih  = (const float*)d_in[2];
    const float* W_hh  = (const float*)d_in[3];
    const float* b_ih  = (const float*)d_in[4];
    const float* b_hh  = (const float*)d_in[5];
    const float* W_out = (const float*)d_in[6];
    const float* b_out = (const float*)d_in[7];
    float* out = (float*)d_out;

    const int T = in_sizes[0];                 // 32768

    float* xg = (float*)d_ws;                  // [T,32] = 4 MB
    float* hs = xg + (size_t)T * NGATES;       // [T, 8] = 1 MB

    // Kernel 1: 4 waves per 128-thread block, 16 timesteps per wave.
    {
        const int waves  = (T + 15) / 16;
        const int blocks = (waves + 3) / 4;
        xgates_wmma_kernel<<<blocks, 128, 0, stream>>>(tok, emb, W_ih, b_ih, b_hh, xg, T);
    }

    // Kernel 2: single-wave serial scan (the fundamental bottleneck).
    lstm_scan_kernel<<<1, 32, 0, stream>>>(xg, W_hh, hs, T);

    // Kernel 3: parallel projection + softmax.
    out_softmax_kernel<<<(T + 255) / 256, 256, 0, stream>>>(hs, W_out, b_out, out, T);
}